// VideoPoseEstimator_47854525612385
// MI455X (gfx1250) — compile-verified
//
#include <hip/hip_runtime.h>
#include <hip/hip_bf16.h>

#define GRID 64
#define HALF 2
#define NPOS 25
#define CCH  256

typedef __attribute__((ext_vector_type(2))) float v2f;
typedef __attribute__((ext_vector_type(8))) float v8f;
typedef __attribute__((ext_vector_type(4))) unsigned int v4u;
typedef __attribute__((ext_vector_type(8))) int v8i;
typedef __attribute__((ext_vector_type(4))) int v4i;

__device__ __forceinline__ float wave_reduce_add(float s) {
    for (int m = 16; m > 0; m >>= 1) s += __shfl_xor(s, m, 32);
    return s;
}

// LDS offset of window position p inside one channel plane [Hd][Wd].
// Invalid / padding positions alias the (always valid) center: their values
// are masked to +inf in the tail, so garbage is fine.
__device__ __forceinline__ int pos_off(int p, int px, int py, int x0, int y0,
                                       int Wd, int tx, int ty) {
    if (p >= NPOS) return ty * Wd + tx;
    const int ix = p % 5, iy = p / 5;
    const int gx = px + ix - HALF, gy = py + iy - HALF;
    const bool valid = (gx >= 0) & (gx < GRID) & (gy >= 0) & (gy < GRID);
    return valid ? ((gy - y0) * Wd + (gx - x0)) : (ty * Wd + tx);
}

__global__ __launch_bounds__(64) void
vpe_track_kernel(const float* __restrict__ keymap,      // (B,256,64,64)
                 const float* __restrict__ first_keys,  // (B,16,256)
                 const float* __restrict__ prev_keys,   // (B,16,256)
                 const int*   __restrict__ prev_kps,    // (B,16,2)
                 float* __restrict__ out,               // [8192 kps][4096 dist]
                 int BK)
{
    __shared__ float sF[CCH * NPOS];   // TDM dest: [c][wy][wx], packed Hd*Wd
    __shared__ float sQf[CCH];
    __shared__ float sQp[CCH];
    __shared__ float sZero[4];
    __shared__ float sNormF2[32];
    __shared__ float sNormQ[2];
    __shared__ float sDotF[32];
    __shared__ float sDotP[32];

    const int bk  = blockIdx.x;
    if (bk >= BK) return;
    const int b   = bk >> 4;
    const int tid = threadIdx.x;
    const int lane = tid & 31;
    const int wave = tid >> 5;

    // Per-block uniform keypoint (force scalar for descriptor build)
    const int px = __builtin_amdgcn_readfirstlane(prev_kps[bk * 2 + 0]);
    const int py = __builtin_amdgcn_readfirstlane(prev_kps[bk * 2 + 1]);

    const int x0 = max(px - HALF, 0);
    const int y0 = max(py - HALF, 0);
    const int Wd = min(px + HALF, GRID - 1) - x0 + 1;
    const int Hd = min(py + HALF, GRID - 1) - y0 + 1;
    const int tx = px - x0, ty = py - y0;
    const int HW = Wd * Hd;

    if (tid < 4) sZero[tid] = 0.0f;
    if (tid < 2) sNormQ[tid] = 0.0f;

    // ---- Tensor Data Mover: fetch clipped Wd x Hd x 128 tile per wave -----
    // 3D tile: dim0=x (contiguous), dim1=y (stride 64), dim2=c (stride 4096).
    {
        const int cbase = wave * 128;
        const unsigned long long gaddr = (unsigned long long)(uintptr_t)
            (keymap + ((size_t)(b * CCH + cbase) * (GRID * GRID))
                    + (size_t)y0 * GRID + x0);
        const unsigned int lds_addr =
            (unsigned int)(uintptr_t)(&sF[0]) + (unsigned int)(cbase * HW) * 4u;

        v4u g0;
        g0[0] = 1u;                                   // count=1, user mode
        g0[1] = lds_addr;                             // LDS dest (bytes)
        g0[2] = (unsigned int)gaddr;                  // global_addr[31:0]
        g0[3] = (unsigned int)((gaddr >> 32) & 0x01FFFFFFull) | 0x80000000u; // type=2

        v8i g1;
        g1[0] = (int)(2u << 16);                      // data_size=4B, mask=0
        g1[1] = (int)(64u << 16);                     // tensor_dim0 = 64
        g1[2] = (int)(64u << 16);                     // tensor_dim1 = 64
        g1[3] = (int)((unsigned int)Wd << 16);        // tile_dim0 = Wd
        g1[4] = (int)((unsigned int)Hd | (128u << 16)); // tile_dim1=Hd, tile_dim2=128
        g1[5] = 64;                                   // tensor_dim0_stride = 64
        g1[6] = (int)(4096u << 16);                   // tensor_dim1_stride = 4096
        g1[7] = 0;

        v4i g2; g2[0] = CCH; g2[1] = 0; g2[2] = 0; g2[3] = 0;  // tensor_dim2
        v4i g3; g3[0] = 0;   g3[1] = 0; g3[2] = 0; g3[3] = 0;
        v8i g4; g4[0] = 0; g4[1] = 0; g4[2] = 0; g4[3] = 0;
                g4[4] = 0; g4[5] = 0; g4[6] = 0; g4[7] = 0;

        __builtin_amdgcn_tensor_load_to_lds(g0, g1, g2, g3, g4, 0);
    }

    // ---- overlap DMA: stage queries, accumulate ||q||^2 -------------------
    {
        const size_t qbase = (size_t)bk * CCH;
        float sf = 0.0f, sp = 0.0f;
        for (int c = tid; c < CCH; c += 64) {
            float qf = first_keys[qbase + c];
            float qp = prev_keys[qbase + c];
            sQf[c] = qf; sQp[c] = qp;
            sf += qf * qf; sp += qp * qp;
        }
        sf = wave_reduce_add(sf);
        sp = wave_reduce_add(sp);
        if (lane == 0) {
            atomicAdd(&sNormQ[0], sf);    // ds_add_f32
            atomicAdd(&sNormQ[1], sp);
        }
    }

    __builtin_amdgcn_s_wait_tensorcnt(0);
    __syncthreads();

    // ---- WMMA phase -------------------------------------------------------
    // acc  = F_tile x [first | prev | 0...]  -> D[p,0]=f.first, D[p,1]=f.prev
    // acc2 = F_tile x F_tile^T               -> diag = ||f_p||^2
    // f32 A(16x4) and B(4x16) fragments are layout-mirrored, so B=A^T reuses
    // the A registers directly.
    {
        const int tile_base = wave << 4;
        const int prow = tile_base + (lane & 15);
        const int koff = (lane < 16) ? 0 : 2;
        const int n    = lane & 15;

        const int offp = pos_off(prow, px, py, x0, y0, Wd, tx, ty);

        // per-lane B pointer: col0 -> first, col1 -> prev, cols 2..15 -> zeros
        const float* qbase = (n == 0) ? sQf : ((n == 1) ? sQp : sZero);
        const int    bstep = (n < 2) ? 1 : 0;
        const float* bptr  = qbase + koff * bstep;

        v8f acc  = {};
        v8f acc2 = {};
#pragma unroll 4
        for (int kk = 0; kk < CCH; kk += 4) {
            v2f a;
            a.x = sF[(kk + koff)     * HW + offp];
            a.y = sF[(kk + koff + 1) * HW + offp];
            v2f bb = *(const v2f*)&bptr[kk * bstep];
            acc  = __builtin_amdgcn_wmma_f32_16x16x4_f32(
                       false, a, false, bb, (short)0, acc,  false, false);
            acc2 = __builtin_amdgcn_wmma_f32_16x16x4_f32(
                       false, a, false, a,  (short)0, acc2, false, false);
        }

        // D layout: VGPR v, lanes 0-15 -> M=v (N=lane), lanes 16-31 -> M=v+8
        if (n <= 1) {
            float* dst = (n == 0) ? sDotF : sDotP;
            const int rowbase = tile_base + ((lane < 16) ? 0 : 8);
#pragma unroll
            for (int v = 0; v < 8; ++v) dst[rowbase + v] = acc[v];
        }
        // diagonal of acc2: lanes 0..7 own (m=lane), lanes 24..31 own (m=lane-16)
        int sel = -1, row = 0;
        if (lane < 8)        { sel = lane;      row = tile_base + lane; }
        else if (lane >= 24) { sel = lane - 24; row = tile_base + 8 + (lane - 24); }
        if (sel >= 0) {
            float dg = acc2[0];
#pragma unroll
            for (int v = 1; v < 8; ++v) dg = (sel == v) ? acc2[v] : dg;
            sNormF2[row] = dg;
        }
    }
    __syncthreads();

    // ---- scalar tail: weight map + 25-way argmin (thread 0) ---------------
    if (tid == 0) {
        const int xl1 = max(tx - 1, 0), xr1 = min(tx + 1, Wd - 1);
        const int yu1 = max(ty - 1, 0), yd1 = min(ty + 1, Hd - 1);
        const int xl2 = max(tx - 2, 0), xr2 = min(tx + 2, Wd - 1);
        const int yu2 = max(ty - 2, 0), yd2 = min(ty + 2, Hd - 1);
        const bool l_skip = (xl1 == 0), r_skip = (xr1 == Wd - 1);
        const bool u_skip = (yu1 == 0), d_skip = (yd1 == Hd - 1);

        const float nQf = sNormQ[0], nQp = sNormQ[1];
        float best = __builtin_inff();
        int bi = 0;
        for (int p = 0; p < NPOS; ++p) {
            const int ix = p % 5, iy = p / 5;
            const int gx = px + ix - HALF, gy = py + iy - HALF;
            const bool valid = (gx >= 0) & (gx < GRID) & (gy >= 0) & (gy < GRID);
            const int wx = gx - x0, wy = gy - y0;

            const bool in_x1 = (wx >= xl1) & (wx <= xr1);
            const bool in_y1 = (wy >= yu1) & (wy <= yd1);
            const bool ring1 = (((wx == xl1) | (wx == xr1)) & in_y1) |
                               (((wy == yu1) | (wy == yd1)) & in_x1);
            float w = ring1 ? 0.6f : 0.5f;
            const bool in_x2 = (wx >= xl2) & (wx <= xr2);
            const bool in_y2 = (wy >= yu2) & (wy <= yd2);
            const bool ring2 = (!l_skip & (wx == xl2) & in_y2) |
                               (!r_skip & (wx == xr2) & in_y2) |
                               (!u_skip & (wy == yu2) & in_x2) |
                               (!d_skip & (wy == yd2) & in_x2);
            if (ring2) w = 0.7f;

            const float nf = sNormF2[p];
            const float df = nf - 2.0f * sDotF[p] + nQf;
            const float dp = nf - 2.0f * sDotP[p] + nQp;
            float d = w * (0.25f * df + 0.75f * dp);
            if (!valid) d = __builtin_inff();
            if (d < best) { best = d; bi = p; }    // first occurrence on ties
        }
        const int ox = bi % 5 - HALF;
        const int oy = bi / 5 - HALF;
        out[bk * 2 + 0]  = (float)(px + ox);
        out[bk * 2 + 1]  = (float)(py + oy);
        out[2 * BK + bk] = best;
    }
}

extern "C" void kernel_launch(void* const* d_in, const int* in_sizes, int n_in,
                              void* d_out, int out_size, void* d_ws, size_t ws_size,
                              hipStream_t stream) {
    (void)in_sizes; (void)n_in; (void)d_ws; (void)ws_size; (void)out_size;
    const float* keymap     = (const float*)d_in[0];
    const float* first_keys = (const float*)d_in[1];
    const float* prev_keys  = (const float*)d_in[2];
    const int*   prev_kps   = (const int*)d_in[3];
    float* out = (float*)d_out;

    const int BK = 256 * 16;
    dim3 grid(BK), block(64);          // 2 waves (wave32) per block
    vpe_track_kernel<<<grid, block, 0, stream>>>(keymap, first_keys, prev_keys,
                                                 prev_kps, out, BK);
}